// CustomModel_59803124630057
// MI455X (gfx1250) — compile-verified
//
#include <hip/hip_runtime.h>

// MI455X / gfx1250 MLP forward:
//   h = relu(x[65536,784] @ W1[784,100] + b1)         (fp32 WMMA 16x16x4, exact f32 math)
//   freq_out = freq_in + colwise count(h[0:B-1] > 0)  (LDS int atomics -> global f32 atomics)
//   out = softmax(h @ W2[100,10] + b2)                (LDS-staged VALU + wave32 shfl_xor)
//
// Roofline: 205.5 MB of x streamed exactly once (TH_LOAD_NT) ~ 9 us @ 23.3 TB/s.
// W1 (313 KB) is repacked once into WMMA B-fragment order in d_ws so the hot loop
// is 1 NT b64 (A) + 7 coalesced b64 (B) + 7 v_wmma per K-step, branch-free.

typedef float v2f __attribute__((ext_vector_type(2)));
typedef float v8f __attribute__((ext_vector_type(8)));

#define NROWS   65536
#define DIN     784
#define DHID    100
#define DOUT    10
#define RPB     128          // rows per block (8 waves x 16-row tiles)
#define HSTRIDE 112          // padded hidden dim (7 tiles of 16)
#define KITERS  (DIN / 4)    // 196
#define NT      7            // N tiles
#define PACK_PAIRS (KITERS * NT * 32)              // b64 pairs in packed W1
#define PACK_BYTES ((size_t)PACK_PAIRS * 8)        // 351,232 B

__global__ __launch_bounds__(128) void freq_init_kernel(const float* __restrict__ fin,
                                                        float* __restrict__ fout) {
    int i = threadIdx.x;
    if (i < DHID) fout[i] = fin[i];
}

// Repack W1[784,100] into B-fragment-contiguous order:
//   frag(kit, t) = 256 B; lane holds {W1[k0*100+col], W1[(k0+1)*100+col]},
//   k0 = kit*4 + 2*(lane>>4), col = t*16 + (lane&15); col >= 100 packed as 0.
__global__ __launch_bounds__(256) void pack_w1_kernel(const float* __restrict__ W1,
                                                      v2f* __restrict__ wp) {
    int tid = blockIdx.x * 256 + threadIdx.x;
    if (tid >= PACK_PAIRS) return;
    const int lane = tid & 31;
    const int t    = (tid >> 5) % NT;
    const int kit  = tid / (32 * NT);
    const int l16  = lane & 15;
    const int h16  = lane >> 4;
    const int k0   = kit * 4 + h16 * 2;
    const int col  = t * 16 + l16;
    v2f b;
    if (col < DHID) {
        b.x = W1[k0 * DHID + col];
        b.y = W1[k0 * DHID + DHID + col];
    } else {
        b.x = 0.0f; b.y = 0.0f;
    }
    wp[tid] = b;
}

template <bool PACKED>
__global__ __launch_bounds__(256) void mlp_fwd_kernel(const float* __restrict__ x,
                                                      const float* __restrict__ W1,
                                                      const v2f*   __restrict__ w1p,
                                                      const float* __restrict__ b1,
                                                      const float* __restrict__ W2,
                                                      const float* __restrict__ b2,
                                                      float* __restrict__ out,
                                                      float* __restrict__ fout) {
    __shared__ float hbuf[RPB * HSTRIDE];   // 57.3 KB staged hidden activations
    __shared__ float w2s[DHID * DOUT];      // 4 KB
    __shared__ float b2s[16];
    __shared__ float b1s[HSTRIDE];
    __shared__ int   freqs[HSTRIDE];

    const int tid = threadIdx.x;

    // ---- cooperative preload of small operands ----
    for (int i = tid; i < DHID * DOUT; i += 256) w2s[i] = W2[i];
    if (tid < 16)      b2s[tid] = (tid < DOUT) ? b2[tid] : 0.0f;
    if (tid < HSTRIDE) { b1s[tid] = (tid < DHID) ? b1[tid] : 0.0f; freqs[tid] = 0; }
    __syncthreads();

    const int lane = tid & 31;        // wave32
    const int wave = tid >> 5;        // 8 waves per block
    const int l16  = lane & 15;
    const int h16  = lane >> 4;       // selects K-pair half (A and B fragments)
    const int rowBase = blockIdx.x * RPB + wave * 16;

    // ---- phase 1: h = relu(x @ W1 + b1) via V_WMMA_F32_16X16X4_F32 ----
    v8f acc[NT] = {};                 // 7 N-tiles of 16 -> cols 0..111 (>=100 are 0)

    // A fragment: lane l (l<16): M=l holds K={0,1}; lane l+16: M=l holds K={2,3}
    const float* xrow = x + (size_t)(rowBase + l16) * DIN;

    for (int kit = 0; kit < KITERS; ++kit) {
        const int k0 = kit * 4 + h16 * 2;
        // x streamed exactly once -> non-temporal keeps W1 hot in WGP$/L2
        v2f a = __builtin_nontemporal_load((const v2f*)(xrow + k0));
#pragma unroll
        for (int t = 0; t < NT; ++t) {
            v2f b;
            if (PACKED) {
                // one coalesced 256 B fragment, identical across all waves
                b = w1p[((size_t)kit * NT + t) * 32 + lane];
            } else {
                // branch-free fallback: clamp address, select zero for padding
                const int col  = t * 16 + l16;
                const int colc = (col < DHID) ? col : (DHID - 1);
                float bx = W1[k0 * DHID + colc];
                float by = W1[k0 * DHID + DHID + colc];
                b.x = (col < DHID) ? bx : 0.0f;
                b.y = (col < DHID) ? by : 0.0f;
            }
            acc[t] = __builtin_amdgcn_wmma_f32_16x16x4_f32(
                false, a, false, b, (short)0, acc[t], false, false);
        }
    }

    // ---- epilogue: bias + relu, freq counts, stage h to LDS ----
#pragma unroll
    for (int t = 0; t < NT; ++t) {
        const int col  = t * 16 + l16;
        const float bias = b1s[col];
        int cnt = 0;
#pragma unroll
        for (int r = 0; r < 8; ++r) {             // C/D layout: M = r + 8*h16, N = l16
            float v = acc[t][r] + bias;
            v = fmaxf(v, 0.0f);
            const int rowl = wave * 16 + r + h16 * 8;
            const int rowg = blockIdx.x * RPB + rowl;
            cnt += (v > 0.0f && rowg != NROWS - 1) ? 1 : 0;   // reference skips last row
            hbuf[rowl * HSTRIDE + col] = v;
        }
        if (cnt) atomicAdd(&freqs[col], cnt);     // ds_add, per-block accumulation
    }
    __syncthreads();

    // ---- phase 2: out = softmax(h @ W2 + b2); 2 lanes per row, 5 cols each ----
    const int rowl = tid >> 1;
    const int half = tid & 1;
    const float* hrow = &hbuf[rowl * HSTRIDE];

    float lg[5];
#pragma unroll
    for (int c = 0; c < 5; ++c) lg[c] = b2s[half * 5 + c];
    for (int k = 0; k < DHID; ++k) {
        const float hk = hrow[k];
#pragma unroll
        for (int c = 0; c < 5; ++c)
            lg[c] = fmaf(hk, w2s[k * DOUT + half * 5 + c], lg[c]);
    }
    float m = lg[0];
#pragma unroll
    for (int c = 1; c < 5; ++c) m = fmaxf(m, lg[c]);
    m = fmaxf(m, __shfl_xor(m, 1, 32));           // pairwise row max (wave32)
    float e[5], s = 0.0f;
#pragma unroll
    for (int c = 0; c < 5; ++c) { e[c] = __expf(lg[c] - m); s += e[c]; }
    s += __shfl_xor(s, 1, 32);                    // pairwise row sum
    const float inv = 1.0f / s;

    const int rowg = blockIdx.x * RPB + rowl;
#pragma unroll
    for (int c = 0; c < 5; ++c)
        __builtin_nontemporal_store(e[c] * inv, &out[rowg * DOUT + half * 5 + c]);

    // ---- freq finalize: exact integer-valued float atomics (order independent) ----
    if (tid < DHID) atomicAdd(&fout[tid], (float)freqs[tid]);
}

extern "C" void kernel_launch(void* const* d_in, const int* in_sizes, int n_in,
                              void* d_out, int out_size, void* d_ws, size_t ws_size,
                              hipStream_t stream) {
    (void)in_sizes; (void)n_in; (void)out_size;
    const float* x    = (const float*)d_in[0];
    const float* W1   = (const float*)d_in[1];
    const float* b1   = (const float*)d_in[2];
    const float* freq = (const float*)d_in[3];
    const float* W2   = (const float*)d_in[4];
    const float* b2   = (const float*)d_in[5];

    float* out  = (float*)d_out;
    float* fout = out + (size_t)NROWS * DOUT;     // outputs concatenated: out, new_freq

    freq_init_kernel<<<1, 128, 0, stream>>>(freq, fout);

    if (ws_size >= PACK_BYTES && d_ws != nullptr) {
        v2f* wp = (v2f*)d_ws;
        pack_w1_kernel<<<(PACK_PAIRS + 255) / 256, 256, 0, stream>>>(W1, wp);
        mlp_fwd_kernel<true><<<NROWS / RPB, 256, 0, stream>>>(x, W1, wp, b1, W2, b2, out, fout);
    } else {
        mlp_fwd_kernel<false><<<NROWS / RPB, 256, 0, stream>>>(x, W1, nullptr, b1, W2, b2, out, fout);
    }
}